// WeightedInverseDistance_9251359556315
// MI455X (gfx1250) — compile-verified
//
#include <hip/hip_runtime.h>

// Weighted inverse-distance interpolation, fused single pass over the N x N
// covar matrix. Gram term of each 16x16 tile via V_WMMA_F32_16X16X4_F32.
//
// d_out layout: [ y_int : N floats ][ covar : N*N floats ] (reference return order)
// d_ws  layout: [ xsp : N*4 floats (scaled coords, K padded to 4) ][ sqn : N floats ]
//
// Roofline: covar write = 268MB @ 23.3TB/s ~= 11.5us floor. Row-normalizer
// (num/den) is fused into the producing pass (no covar re-read). Each wave
// completes full 128B lines (32 consecutive columns) per iteration so NT
// stores never evict half-written lines.

#define IDW_EPS 1e-6f

typedef __attribute__((ext_vector_type(2))) float v2f;
typedef __attribute__((ext_vector_type(8))) float v8f;

__global__ __launch_bounds__(256) void idw_prep(const float* __restrict__ x,
                                                const float* __restrict__ ls,
                                                float* __restrict__ xsp,
                                                float* __restrict__ sqn, int N) {
  int i = blockIdx.x * blockDim.x + threadIdx.x;
  if (i >= N) return;
  float a0 = x[3 * i + 0] / ls[0];
  float a1 = x[3 * i + 1] / ls[1];
  float a2 = x[3 * i + 2] / ls[2];
  float4 v = make_float4(a0, a1, a2, 0.0f);  // pad K: 3 -> 4 for 16x16x4 WMMA
  *(float4*)(xsp + (size_t)4 * i) = v;
  sqn[i] = a0 * a0 + a1 * a1 + a2 * a2;
}

// One block = 16 output rows (I..I+15), 8 waves. Wave w owns the 32-column
// stripes starting at (w + 8*t)*32: two adjacent 16x16 tiles per iteration
// -> two independent WMMAs, and every 128B output line completed at once.
__global__ __launch_bounds__(256) void idw_main(const float* __restrict__ xsp,
                                                const float* __restrict__ sqn,
                                                const float* __restrict__ y,
                                                const float* __restrict__ power,
                                                float* __restrict__ yint,
                                                float* __restrict__ covar, int N) {
  __shared__ float2 s_col[8192];       // {sqn[j], y[j]} per column, 64KB
  __shared__ float  s_red[8][16][2];   // [wave][row][num,den]

  const int tid = threadIdx.x;
  for (int t = tid; t < N; t += 256) s_col[t] = make_float2(sqn[t], y[t]);
  __syncthreads();

  const int lane = tid & 31;
  const int wave = tid >> 5;
  const int half = lane >> 4;   // 0: K=0,1 / rows m+0 ; 1: K=2,3 / rows m+8
  const int l16  = lane & 15;

  const int   I = blockIdx.x * 16;
  const float p = power[0];

  // A-matrix 16x4 f32 layout: lanes 0-15 {K0,K1}, lanes 16-31 {K2,K3}, M = lane%16
  v2f a = *(const v2f*)(xsp + (size_t)(I + l16) * 4 + half * 2);

  float  sqrow[8];
  size_t rowoff[8];
#pragma unroll
  for (int r = 0; r < 8; ++r) {
    sqrow[r]  = sqn[I + 8 * half + r];
    rowoff[r] = (size_t)(I + 8 * half + r) * N;
  }

  float num[8], den[8];
#pragma unroll
  for (int r = 0; r < 8; ++r) { num[r] = 0.0f; den[r] = 0.0f; }

  const int nstripes = N >> 5;                  // 32-column stripes
  for (int St = wave; St < nstripes; St += 8) {
    const int J = St << 5;
    // Two adjacent B tiles (4x16 f32, mirrored A layout)
    v2f b0 = *(const v2f*)(xsp + (size_t)(J + l16) * 4 + half * 2);
    v2f b1 = *(const v2f*)(xsp + (size_t)(J + 16 + l16) * 4 + half * 2);
    if (St + 8 < nstripes)  // prefetch next stripe's B rows (256 rows ahead)
      __builtin_prefetch(xsp + (size_t)(J + 256 + l16) * 4, 0, 1);

    v8f c0 = {}, c1 = {};
    c0 = __builtin_amdgcn_wmma_f32_16x16x4_f32(false, a, false, b0,
                                               (short)0, c0, false, false);
    c1 = __builtin_amdgcn_wmma_f32_16x16x4_f32(false, a, false, b1,
                                               (short)0, c1, false, false);

    const float2 cy0 = s_col[J + l16];        // {sqn, y} for col J+l16
    const float2 cy1 = s_col[J + 16 + l16];   // {sqn, y} for col J+16+l16

    // Staged computation: 16 independent elements per trans stage so the
    // v_sqrt/v_log/v_exp pipelines fill instead of chaining serially.
    float d2[16], dd[16], lg[16], kk[16];
#pragma unroll
    for (int r = 0; r < 8; ++r) {
      d2[r]     = sqrow[r] + cy0.x - 2.0f * c0[r];
      d2[r + 8] = sqrow[r] + cy1.x - 2.0f * c1[r];
    }
#pragma unroll
    for (int e = 0; e < 16; ++e)
      dd[e] = __builtin_amdgcn_sqrtf(__builtin_fmaxf(d2[e], 0.0f)) + IDW_EPS;
#pragma unroll
    for (int e = 0; e < 16; ++e)
      lg[e] = __builtin_amdgcn_logf(dd[e]);        // native log2
#pragma unroll
    for (int e = 0; e < 16; ++e)
      kk[e] = __builtin_amdgcn_exp2f(-p * lg[e]);  // (d+eps)^(-p)
#pragma unroll
    for (int r = 0; r < 8; ++r) {
      // streaming NT stores: this wave writes both 64B halves of each 128B line
      __builtin_nontemporal_store(kk[r],     covar + rowoff[r] + J + l16);
      __builtin_nontemporal_store(kk[r + 8], covar + rowoff[r] + J + 16 + l16);
      num[r] += kk[r] * cy0.y + kk[r + 8] * cy1.y;
      den[r] += kk[r] + kk[r + 8];
    }
  }

  // Reduce the 16 column-lanes of each half (xor tree stays inside each half)
#pragma unroll
  for (int r = 0; r < 8; ++r) {
#pragma unroll
    for (int m = 8; m >= 1; m >>= 1) {
      num[r] += __shfl_xor(num[r], m, 32);
      den[r] += __shfl_xor(den[r], m, 32);
    }
  }
  if (l16 == 0) {
#pragma unroll
    for (int r = 0; r < 8; ++r) {
      s_red[wave][r + 8 * half][0] = num[r];
      s_red[wave][r + 8 * half][1] = den[r];
    }
  }
  __syncthreads();
  if (tid < 16) {  // deterministic cross-wave reduction + final normalize
    float n = 0.0f, d = 0.0f;
#pragma unroll
    for (int w = 0; w < 8; ++w) { n += s_red[w][tid][0]; d += s_red[w][tid][1]; }
    yint[I + tid] = n / d;
  }
}

extern "C" void kernel_launch(void* const* d_in, const int* in_sizes, int n_in,
                              void* d_out, int out_size, void* d_ws, size_t ws_size,
                              hipStream_t stream) {
  const float* x     = (const float*)d_in[0];  // [N,3]
  const float* y     = (const float*)d_in[1];  // [N,1]
  const float* ls    = (const float*)d_in[2];  // [3]
  const float* power = (const float*)d_in[3];  // [1]
  const int N = in_sizes[1];                   // 8192

  float* xsp = (float*)d_ws;                   // N*4 floats
  float* sqn = xsp + (size_t)N * 4;            // N floats

  float* yint  = (float*)d_out;                // N
  float* covar = yint + N;                     // N*N

  idw_prep<<<(N + 255) / 256, 256, 0, stream>>>(x, ls, xsp, sqn, N);
  idw_main<<<N / 16, 256, 0, stream>>>(xsp, sqn, y, power, yint, covar, N);
}